// DSS_67637144978496
// MI455X (gfx1250) — compile-verified
//
#include <hip/hip_runtime.h>

typedef __attribute__((ext_vector_type(2))) float v2f;
typedef __attribute__((ext_vector_type(8))) float v8f;

#define N_STATE 64          // kernel specialized for N=64 (2 n's per K=4 step x 32 steps)
#define WAVES   8           // waves per block (wave32)
#define BLOCK   (32 * WAVES)
#define TILE_L  256         // 16 (m rows) x 16 (j cols) output l-values per WMMA tile
#define DSS_EPS 1e-7f

// K[c,h, tile*256 + 16m + j] = Re( sum_n A_c[m,n] * z[n]^j ),  z = exp(dt_Lambda)
// A_c[m,n] = (Wc[n] * z[n]^{16m}) * exp(dt_Lambda[n]*(tile*256 - shift[n]))
//          =  af[n][m]            * g[n](tile)
// Real-ified as a 16x16x128 f32 matmul: A' = [re,im], B' = [re,-im] interleaved.
// af/bf are block-shared tables resident in VGPRs; g is 64 complex values per wave,
// recomputed cooperatively in LDS every tile iteration (can't be register-hoisted).
__global__ __launch_bounds__(BLOCK)
void dss_wmma_kernel(const float* __restrict__ Lambda,   // [N,2]
                     const float* __restrict__ log_dt,   // [H,2]
                     const float* __restrict__ W,        // [C,H,N,2]
                     float* __restrict__ out,            // [C,H,L]
                     int H, int L)
{
    __shared__ float s_dre[N_STATE], s_dli[N_STATE];     // dt_Lambda re/im
    __shared__ float s_wcr[N_STATE], s_wci[N_STATE];     // Wc re/im
    __shared__ float s_shift[N_STATE];                   // Pmax phase shift (l units)
    __shared__ v2f   s_bf[N_STATE * 16];                 // [n][j] = (Re z^j, -Im z^j)
    __shared__ v2f   s_af[N_STATE * 16];                 // [n][j] = Wc[n] * z[n]^(16j)
    __shared__ v2f   s_g [WAVES * N_STATE];              // [w][n] = exp(dtL*(tile_w*256-shift))

    const int tid  = threadIdx.x;
    const int ch   = blockIdx.x;          // c*H + h
    const int h    = ch % H;
    const int wave = tid >> 5;
    const int lane = tid & 31;
    const int hl   = lane >> 4;           // lane half: selects n parity
    const int mj   = lane & 15;           // A-row m / B-col j
    const float fL = (float)L;

    // ---- phase 1: per-n quantities, one thread per n (block-shared) ----
    if (tid < N_STATE) {
        const int n = tid;
        const float dt_re = expf(log_dt[2 * h + 0]);
        const float dt_im = expf(log_dt[2 * h + 1]);
        const float lre = Lambda[2 * n + 0];
        const float lim = Lambda[2 * n + 1];
        const float dre = dt_re * lre;                 // Re dt_Lambda
        const float dli = dt_im * lim;                 // Im dt_Lambda
        const bool  gt0 = (lre > 0.0f);
        const float sgn = gt0 ? -1.0f : 1.0f;

        // num = exp(sgn*dtL) - 1
        const float er    = expf(sgn * dre);
        const float num_r = er * cosf(sgn * dli) - 1.0f;
        const float num_i = er * sinf(sgn * dli);
        // den = exp(L*sgn*dtL) - 1
        const float eL    = expf(fL * sgn * dre);
        const float aL    = fL * sgn * dli;
        const float den_r = eL * cosf(aL) - 1.0f;
        const float den_i = eL * sinf(aL);
        // x = den * Lambda ; recip = conj(x)/(|x|^2 + eps)
        const float xr  = den_r * lre - den_i * lim;
        const float xi  = den_r * lim + den_i * lre;
        const float inv = 1.0f / (xr * xr + xi * xi + DSS_EPS);
        const float rr  =  xr * inv;
        const float ri  = -xi * inv;
        // Wc = (W_re + i W_im) * num * recip
        const float wr  = W[(ch * N_STATE + n) * 2 + 0];
        const float wi  = W[(ch * N_STATE + n) * 2 + 1];
        const float t1r = wr * num_r - wi * num_i;
        const float t1i = wr * num_i + wi * num_r;

        s_dre[n]   = dre;
        s_dli[n]   = dli;
        s_wcr[n]   = t1r * rr - t1i * ri;
        s_wci[n]   = t1r * ri + t1i * rr;
        s_shift[n] = gt0 ? (fL - 1.0f) : 0.0f;
    }
    __syncthreads();

    // ---- phase 1b: fill block-shared tables bf[n][j]=z^j', af[n][j]=Wc*z^(16j) ----
#pragma unroll
    for (int q = 0; q < (N_STATE * 16) / BLOCK; ++q) {   // 1024/256 = 4
        const int e = tid + q * BLOCK;
        const int n = e >> 4;
        const int j = e & 15;
        const float dre = s_dre[n], dli = s_dli[n];

        const float eb = expf(dre * (float)j);
        s_bf[e] = (v2f){ eb * cosf(dli * (float)j), -eb * sinf(dli * (float)j) };

        const float e16 = expf(dre * (float)(16 * j));
        const float c16 = cosf(dli * (float)(16 * j));
        const float s16 = sinf(dli * (float)(16 * j));
        const float wcr = s_wcr[n], wci = s_wci[n];
        s_af[e] = (v2f){ e16 * (wcr * c16 - wci * s16),
                         e16 * (wcr * s16 + wci * c16) };
    }
    __syncthreads();

    // ---- resident fragments: only af + bf (128 VGPRs) + accumulators ----
    v2f bfr[32], afr[32];
#pragma unroll
    for (int t = 0; t < 32; ++t) {
        const int e = (2 * t + hl) * 16 + mj;
        bfr[t] = s_bf[e];
        afr[t] = s_af[e];
    }

    const int ntiles = (L + TILE_L - 1) / TILE_L;
    const int iters  = (ntiles + WAVES - 1) / WAVES;     // uniform across all waves
    float* __restrict__ outh = out + (size_t)ch * (size_t)L;

    for (int itr = 0; itr < iters; ++itr) {
        __syncthreads();   // previous iteration's s_g reads complete before rewrite

        // cooperative per-(wave,n) tile factors: 512 entries, 2 per thread
#pragma unroll
        for (int q = 0; q < (WAVES * N_STATE) / BLOCK; ++q) {   // 2
            const int idx = tid + q * BLOCK;
            const int w   = idx >> 6;
            const int n   = idx & (N_STATE - 1);
            const int tw  = w + itr * WAVES;
            const float phg = (float)(tw * TILE_L) - s_shift[n];
            const float em  = expf(s_dre[n] * phg);
            s_g[idx] = (v2f){ em * cosf(s_dli[n] * phg), em * sinf(s_dli[n] * phg) };
        }
        __syncthreads();

        const int tile = wave + itr * WAVES;
        v8f acc0 = {0.f, 0.f, 0.f, 0.f, 0.f, 0.f, 0.f, 0.f};
        v8f acc1 = acc0, acc2 = acc0, acc3 = acc0;

#pragma unroll
        for (int t = 0; t < 32; t += 4) {
            v2f a0, a1, a2, a3;
            {   const v2f g = s_g[wave * N_STATE + 2 * (t + 0) + hl];  const v2f f = afr[t + 0];
                a0 = (v2f){ f.x * g.x - f.y * g.y, f.x * g.y + f.y * g.x }; }
            {   const v2f g = s_g[wave * N_STATE + 2 * (t + 1) + hl];  const v2f f = afr[t + 1];
                a1 = (v2f){ f.x * g.x - f.y * g.y, f.x * g.y + f.y * g.x }; }
            {   const v2f g = s_g[wave * N_STATE + 2 * (t + 2) + hl];  const v2f f = afr[t + 2];
                a2 = (v2f){ f.x * g.x - f.y * g.y, f.x * g.y + f.y * g.x }; }
            {   const v2f g = s_g[wave * N_STATE + 2 * (t + 3) + hl];  const v2f f = afr[t + 3];
                a3 = (v2f){ f.x * g.x - f.y * g.y, f.x * g.y + f.y * g.x }; }
            acc0 = __builtin_amdgcn_wmma_f32_16x16x4_f32(false, a0, false, bfr[t + 0],
                                                         (short)0, acc0, false, false);
            acc1 = __builtin_amdgcn_wmma_f32_16x16x4_f32(false, a1, false, bfr[t + 1],
                                                         (short)0, acc1, false, false);
            acc2 = __builtin_amdgcn_wmma_f32_16x16x4_f32(false, a2, false, bfr[t + 2],
                                                         (short)0, acc2, false, false);
            acc3 = __builtin_amdgcn_wmma_f32_16x16x4_f32(false, a3, false, bfr[t + 3],
                                                         (short)0, acc3, false, false);
        }
        const v8f acc = (acc0 + acc1) + (acc2 + acc3);

        // D layout: element r -> M = r + 8*hl, N = mj ; l = tile*256 + 16*M + N
        const int lbase = tile * TILE_L + 16 * 8 * hl + mj;
#pragma unroll
        for (int r = 0; r < 8; ++r) {
            const int l = lbase + 16 * r;
            if (l < L) outh[l] = acc[r];     // also guards tail tiles (tile >= ntiles)
        }
    }
}

extern "C" void kernel_launch(void* const* d_in, const int* in_sizes, int n_in,
                              void* d_out, int out_size, void* d_ws, size_t ws_size,
                              hipStream_t stream) {
    // inputs: [0]=L (scalar int, derived on host from sizes), [1]=Lambda [N,2],
    //         [2]=log_dt [H,2], [3]=W [C,H,N,2] ; output: K [C,H,L] f32
    const float* Lambda = (const float*)d_in[1];
    const float* log_dt = (const float*)d_in[2];
    const float* W      = (const float*)d_in[3];
    float* out          = (float*)d_out;

    const int N = in_sizes[1] / 2;                 // 64
    const int H = in_sizes[2] / 2;                 // 128
    const int C = in_sizes[3] / (H * N * 2);       // 1
    const int L = out_size / (C * H);              // 8192
    (void)N; (void)n_in; (void)d_ws; (void)ws_size;

    dim3 grid(C * H);
    dim3 block(BLOCK);
    hipLaunchKernelGGL(dss_wmma_kernel, grid, block, 0, stream,
                       Lambda, log_dt, W, out, H, L);
}